// PointHead_31945966747963
// MI455X (gfx1250) — compile-verified
//
#include <hip/hip_runtime.h>

typedef __attribute__((ext_vector_type(2))) float v2f;
typedef __attribute__((ext_vector_type(8))) float v8f;

#define B_SZ 16
#define NPTS 1024
#define NIMP 768
#define NOVER 7168          // 7*1024
#define NSORT 8192
#define KPAD 136            // 135 padded to multiple of 4
#define HS   264            // 256 + pad
#define REND_ELEMS (16*7*1024)

// ---------------- Kernel 1: uncertainty keys -------------------------------
// key = (ordered_bits(unc) << 32) | (0xFFFFFFFF - idx)  -> one descending u64
// sort reproduces jax.lax.top_k ordering (value desc, ties: lower idx first).
__global__ __launch_bounds__(256) void k_unc(const float* __restrict__ coarse,
                                             const float* __restrict__ over_gen,
                                             unsigned long long* __restrict__ keys) {
    int gid = blockIdx.x * 256 + threadIdx.x;
    if (gid >= B_SZ * NOVER) return;
    int b = gid / NOVER;
    unsigned int i = (unsigned int)(gid - b * NOVER);

    float px = over_gen[(size_t)gid * 2 + 0];
    float py = over_gen[(size_t)gid * 2 + 1];
    // grid_sample(align_corners=False): x = px*W - 0.5
    float x = px * 64.0f - 0.5f, y = py * 64.0f - 0.5f;
    float fx0 = floorf(x), fy0 = floorf(y);
    float wx = x - fx0, wy = y - fy0;
    int ix0 = (int)fx0, iy0 = (int)fy0;

    const float* cb = coarse + (size_t)b * 7 * 4096;
    float s0 = 0.0f, s1 = 0.0f;
    #pragma unroll
    for (int dy = 0; dy < 2; ++dy) {
        #pragma unroll
        for (int dx = 0; dx < 2; ++dx) {
            int ix = ix0 + dx, iy = iy0 + dy;
            if (ix >= 0 && ix < 64 && iy >= 0 && iy < 64) {
                float t1 = -1e30f, t2 = -1e30f;
                #pragma unroll
                for (int ch = 0; ch < 7; ++ch) {
                    float v = cb[ch * 4096 + iy * 64 + ix];
                    if (v > t1) { t2 = t1; t1 = v; }
                    else if (v > t2) { t2 = v; }
                }
                float wgt = (dx ? wx : 1.0f - wx) * (dy ? wy : 1.0f - wy);
                s0 += t1 * wgt;
                s1 += t2 * wgt;
            }
        }
    }
    float unc = s1 - s0;
    unsigned int fb = __float_as_uint(unc);
    unsigned int ord = (fb & 0x80000000u) ? ~fb : (fb | 0x80000000u);
    keys[gid] = ((unsigned long long)ord << 32) |
                (unsigned long long)(0xFFFFFFFFu - i);
}

// ---------------- Kernel 2: per-batch bitonic top-k + emit points ----------
__global__ __launch_bounds__(1024) void k_sort(const unsigned long long* __restrict__ keys,
                                               const float* __restrict__ over_gen,
                                               const float* __restrict__ coverage,
                                               float* __restrict__ points_out) {
    __shared__ unsigned long long s[NSORT];
    int b = blockIdx.x;
    int tid = threadIdx.x;

    for (int i = tid; i < NSORT; i += 1024)
        s[i] = (i < NOVER) ? keys[(size_t)b * NOVER + i] : 0ULL;
    __syncthreads();

    for (int k = 2; k <= NSORT; k <<= 1) {
        for (int j = k >> 1; j > 0; j >>= 1) {
            for (int t = 0; t < NSORT; t += 1024) {
                int i = t + tid;
                int ixj = i ^ j;
                if (ixj > i) {
                    unsigned long long a = s[i], c = s[ixj];
                    bool descBlock = ((i & k) == 0);
                    bool doSwap = descBlock ? (a < c) : (a > c);
                    if (doSwap) { s[i] = c; s[ixj] = a; }
                }
            }
            __syncthreads();
        }
    }

    if (tid < NIMP) {
        unsigned int idx = 0xFFFFFFFFu - (unsigned int)(s[tid] & 0xFFFFFFFFULL);
        points_out[(size_t)b * NPTS * 2 + tid * 2 + 0] = over_gen[(size_t)b * NOVER * 2 + (size_t)idx * 2 + 0];
        points_out[(size_t)b * NPTS * 2 + tid * 2 + 1] = over_gen[(size_t)b * NOVER * 2 + (size_t)idx * 2 + 1];
    } else {
        int ci = tid - NIMP; // 0..255
        points_out[(size_t)b * NPTS * 2 + tid * 2 + 0] = coverage[(size_t)b * 256 * 2 + ci * 2 + 0];
        points_out[(size_t)b * NPTS * 2 + tid * 2 + 1] = coverage[(size_t)b * 256 * 2 + ci * 2 + 1];
    }
}

// ---------------- Kernel 3: fused sampling + 2-layer MLP via f32 WMMA ------
// grid = 16 batches * 64 tiles of 16 points; block = 128 threads (4 waves).
__global__ __launch_bounds__(128) void k_mlp(const float* __restrict__ coarse,
                                             const float* __restrict__ fine,
                                             const float* __restrict__ w1,
                                             const float* __restrict__ b1,
                                             const float* __restrict__ w2,
                                             const float* __restrict__ b2,
                                             const float* __restrict__ points,
                                             float* __restrict__ rend) {
    __shared__ float featA[16 * KPAD];   // A: 16 points x 136 (K padded, col 135 = 0)
    __shared__ float hL[16 * HS];        // hidden: 16 x 256 (+pad)
    __shared__ float spx[16], spy[16];

    int b    = blockIdx.x >> 6;
    int tile = blockIdx.x & 63;
    int m0   = tile * 16;
    int tid  = threadIdx.x;

    if (tid < 16) {
        spx[tid] = points[(size_t)b * NPTS * 2 + (m0 + tid) * 2 + 0];
        spy[tid] = points[(size_t)b * NPTS * 2 + (m0 + tid) * 2 + 1];
    }
    for (int i = tid; i < 16 * KPAD; i += 128) featA[i] = 0.0f;
    __syncthreads();

    // ---- sample feat: 16 points x 135 channels (7 coarse + 128 fine) ----
    for (int task = tid; task < 16 * 135; task += 128) {
        int m  = task / 135;
        int ch = task - m * 135;
        const float* base; int H, W;
        if (ch < 7) { base = coarse + ((size_t)b * 7 + ch) * 4096;          H = 64;  W = 64;  }
        else        { base = fine   + ((size_t)b * 128 + (ch - 7)) * 65536; H = 256; W = 256; }
        float x = spx[m] * (float)W - 0.5f;
        float y = spy[m] * (float)H - 0.5f;
        float fx0 = floorf(x), fy0 = floorf(y);
        float wx = x - fx0, wy = y - fy0;
        int ix0 = (int)fx0, iy0 = (int)fy0;
        float acc = 0.0f;
        #pragma unroll
        for (int dy = 0; dy < 2; ++dy) {
            #pragma unroll
            for (int dx = 0; dx < 2; ++dx) {
                int ix = ix0 + dx, iy = iy0 + dy;
                if (ix >= 0 && ix < W && iy >= 0 && iy < H) {
                    float wgt = (dx ? wx : 1.0f - wx) * (dy ? wy : 1.0f - wy);
                    acc += base[iy * W + ix] * wgt;
                }
            }
        }
        featA[m * KPAD + ch] = acc;
    }
    __syncthreads();

    int wave  = tid >> 5;
    int lane  = tid & 31;
    int mfrag = lane & 15;
    int hi    = lane >> 4;       // 0: lanes 0-15, 1: lanes 16-31
    int koff  = hi * 2;

    // ---- GEMM1: feat[16x135] @ w1[135x256] + b1, relu -> hL[16x256] ----
    // each wave owns 64 output columns (4 N-tiles of 16)
    for (int nt = 0; nt < 4; ++nt) {
        int n0   = wave * 64 + nt * 16;
        int ncol = n0 + mfrag;
        v8f acc = {};
        // kt = 0..32: max row index is 32*4+2+1 = 131 < 135 -> no guards
        for (int kt = 0; kt < 33; ++kt) {
            int kk = kt * 4 + koff;
            v2f a, bb;
            a.x  = featA[mfrag * KPAD + kk];
            a.y  = featA[mfrag * KPAD + kk + 1];
            bb.x = w1[kk * 256 + ncol];
            bb.y = w1[(kk + 1) * 256 + ncol];
            acc = __builtin_amdgcn_wmma_f32_16x16x4_f32(
                false, a, false, bb, (short)0, acc, false, false);
        }
        // tail kt = 33: rows 132..135; row 135 masked (clamped addr, x0 weight)
        {
            int kk = 132 + koff;                 // 132 (lo half) or 134 (hi half)
            int kn = (kk + 1 < 135) ? (kk + 1) : 134;
            float maskY = (kk + 1 < 135) ? 1.0f : 0.0f;
            v2f a, bb;
            a.x  = featA[mfrag * KPAD + kk];
            a.y  = featA[mfrag * KPAD + kk + 1]; // col 135 is zero in LDS
            bb.x = w1[kk * 256 + ncol];
            bb.y = w1[kn * 256 + ncol] * maskY;
            acc = __builtin_amdgcn_wmma_f32_16x16x4_f32(
                false, a, false, bb, (short)0, acc, false, false);
        }
        float bias = b1[ncol];
        #pragma unroll
        for (int r = 0; r < 8; ++r) {
            int M = r + 8 * hi;
            float hv = acc[r] + bias;
            hL[M * HS + ncol] = hv > 0.0f ? hv : 0.0f;
        }
    }
    __syncthreads();

    // ---- GEMM2: hL[16x256] @ w2[256x7] + b2 -> rend (wave 0 only) ----
    if (wave == 0) {
        int ncol  = mfrag;                      // output channel, valid if < 7
        int ncl   = (ncol < 7) ? ncol : 6;      // clamped (always in-bounds)
        float nmk = (ncol < 7) ? 1.0f : 0.0f;   // mask for padded columns
        v8f acc = {};
        for (int kt = 0; kt < 64; ++kt) {
            int kk = kt * 4 + koff;
            v2f a, bb;
            a.x  = hL[mfrag * HS + kk];
            a.y  = hL[mfrag * HS + kk + 1];
            bb.x = w2[kk * 7 + ncl] * nmk;
            bb.y = w2[(kk + 1) * 7 + ncl] * nmk;
            acc = __builtin_amdgcn_wmma_f32_16x16x4_f32(
                false, a, false, bb, (short)0, acc, false, false);
        }
        if (ncol < 7) {
            float bias = b2[ncol];
            #pragma unroll
            for (int r = 0; r < 8; ++r) {
                int M = r + 8 * hi;
                // rend layout: [B, 7, 1024]
                rend[(size_t)b * 7168 + ncol * 1024 + (m0 + M)] = acc[r] + bias;
            }
        }
    }
}

extern "C" void kernel_launch(void* const* d_in, const int* in_sizes, int n_in,
                              void* d_out, int out_size, void* d_ws, size_t ws_size,
                              hipStream_t stream) {
    const float* fine     = (const float*)d_in[0];
    const float* coarse   = (const float*)d_in[1];
    const float* w1       = (const float*)d_in[2];
    const float* b1       = (const float*)d_in[3];
    const float* w2       = (const float*)d_in[4];
    const float* b2       = (const float*)d_in[5];
    const float* over_gen = (const float*)d_in[6];
    const float* coverage = (const float*)d_in[7];

    float* out    = (float*)d_out;
    float* rend   = out;                 // [16,7,1024]
    float* points = out + REND_ELEMS;    // [16,1024,2]
    unsigned long long* keys = (unsigned long long*)d_ws;  // 16*7168*8 B

    k_unc <<<(B_SZ * NOVER + 255) / 256, 256, 0, stream>>>(coarse, over_gen, keys);
    k_sort<<<B_SZ, 1024, 0, stream>>>(keys, over_gen, coverage, points);
    k_mlp <<<B_SZ * 64, 128, 0, stream>>>(coarse, fine, w1, b1, w2, b2, points, rend);
}